// ThreeBodyEquiGraphConvSimple_39651138076971
// MI455X (gfx1250) — compile-verified
//
#include <hip/hip_runtime.h>
#include <hip/hip_bf16.h>
#include <math.h>

typedef __attribute__((ext_vector_type(2))) float v2f;
typedef __attribute__((ext_vector_type(8))) float v8f;

#define DEVINL __device__ __forceinline__

namespace {
constexpr int kN = 50000;
constexpr int kE = 800000;
constexpr int kNodeTiles = kN / 16;   // 3125
constexpr int kEdgeTiles = kE / 16;   // 50000
constexpr float kPiOverCut = 0.31415926535897931f;  // pi / 10
}

// One K=4 step of D = A(16x4) * B(4x16) + C, f32 WMMA.
// A layout (ISA 7.12.2): lane holds row (lane&15), contiguous col pair at ka+2*(lane>>4).
// B layout: v0 = W[kb+2*hi][n], v1 = W[kb+2*hi+1][n], n = lane&15.
DEVINL v8f wmma_ab(const float* arow, int ka, const float* __restrict__ W,
                   int ldw, int kb, int lane, v8f c) {
  int hi = lane >> 4;
  int n  = lane & 15;
  v2f a = *(const v2f*)(arow + ka + 2 * hi);
  int row = kb + 2 * hi;
  v2f b;
  b.x = W[row * ldw + n];
  b.y = W[(row + 1) * ldw + n];
  return __builtin_amdgcn_wmma_f32_16x16x4_f32(false, a, false, b, (short)0, c,
                                               false, false);
}

// Stage a C-layout 16x16 tile into LDS (row stride 18 -> bank-conflict free)
// so it can be re-read in A layout. DS ops are in-order within a wave; the
// s_wait_dscnt + memory clobber keeps the compiler from reordering.
DEVINL void c_to_lds(float* sm, v8f c, int lane) {
  int hi = lane >> 4;
  int n  = lane & 15;
#pragma unroll
  for (int r = 0; r < 8; ++r) sm[(r + 8 * hi) * 18 + n] = c[r];
  asm volatile("s_wait_dscnt 0" ::: "memory");
}

// Fast math: single v_rcp_f32 instead of the IEEE division expansion.
DEVINL float frcp(float x) { return __builtin_amdgcn_rcpf(x); }
DEVINL float silu1(float x) { return x * frcp(1.f + __expf(-x)); }
DEVINL float sigm1(float x) { return frcp(1.f + __expf(-x)); }

// ---------------------------------------------------------------------------
// Kernel 1: ns = silu(node_s @ Wns + bns); nv = node_v @ Wnv; nv_norm
// ---------------------------------------------------------------------------
__global__ void node_in_kernel(const float* __restrict__ node_s,
                               const float* __restrict__ node_v,
                               const float* __restrict__ Wns,
                               const float* __restrict__ bns,
                               const float* __restrict__ Wnv,
                               float* __restrict__ ns, float* __restrict__ nv,
                               float* __restrict__ nvn) {
  int wave = threadIdx.x >> 5;
  int lane = threadIdx.x & 31;
  int tile = blockIdx.x * (blockDim.x >> 5) + wave;
  if (tile >= kNodeTiles) return;
  int base = tile * 16;
  int n = lane & 15, hi = lane >> 4;

  // ns tile
  v8f c;
  float bb = bns[n];
#pragma unroll
  for (int r = 0; r < 8; ++r) c[r] = bb;
  const float* arow = node_s + (size_t)(base + n) * 128;
#pragma unroll 8
  for (int k0 = 0; k0 < 128; k0 += 4) c = wmma_ab(arow, k0, Wns, 16, k0, lane, c);
#pragma unroll
  for (int r = 0; r < 8; ++r) c[r] = silu1(c[r]);
#pragma unroll
  for (int r = 0; r < 8; ++r) ns[(size_t)(base + r + 8 * hi) * 16 + n] = c[r];

  // nv tiles (3 spatial dims) + squared-norm accumulation
  float ss[8];
#pragma unroll
  for (int r = 0; r < 8; ++r) ss[r] = 1e-12f;
#pragma unroll
  for (int d = 0; d < 3; ++d) {
    v8f cv;
#pragma unroll
    for (int r = 0; r < 8; ++r) cv[r] = 0.f;
    const float* avrow = node_v + ((size_t)(base + n) * 3 + d) * 128;
#pragma unroll 8
    for (int k0 = 0; k0 < 128; k0 += 4)
      cv = wmma_ab(avrow, k0, Wnv, 16, k0, lane, cv);
#pragma unroll
    for (int r = 0; r < 8; ++r) {
      nv[((size_t)(base + r + 8 * hi) * 3 + d) * 16 + n] = cv[r];
      ss[r] += cv[r] * cv[r];
    }
  }
#pragma unroll
  for (int r = 0; r < 8; ++r)
    nvn[(size_t)(base + r + 8 * hi) * 16 + n] = sqrtf(ss[r]);
}

// ---------------------------------------------------------------------------
// Kernel 2: full edge MLP + residual edge outputs + scatter-add to nodes
// ---------------------------------------------------------------------------
__global__ void edge_kernel(
    const float* __restrict__ edge_s, const float* __restrict__ edge_v,
    const float* __restrict__ dist, const float* __restrict__ vctr,
    const int* __restrict__ src, const int* __restrict__ dst,
    const float* __restrict__ Wen, const float* __restrict__ ben,
    const float* __restrict__ Wtp, const float* __restrict__ btp,
    const float* __restrict__ Wg1, const float* __restrict__ bg1,
    const float* __restrict__ Wg2, const float* __restrict__ bg2,
    const float* __restrict__ Wtv, const float* __restrict__ btv,
    const float* __restrict__ ns, const float* __restrict__ nv,
    float* __restrict__ n_es, float* __restrict__ n_ev,
    float* __restrict__ es_out, float* __restrict__ ev_out) {
  __shared__ float smem[8][16 * 18];
  int wave = threadIdx.x >> 5;
  int lane = threadIdx.x & 31;
  int tile = blockIdx.x * (blockDim.x >> 5) + wave;
  if (tile >= kEdgeTiles) return;
  int base = tile * 16;
  int n = lane & 15, hi = lane >> 4;
  float* sm = smem[wave];

  // tp = edge_s @ Wtp + btp
  v8f ctp;
  float bb = btp[n];
#pragma unroll
  for (int r = 0; r < 8; ++r) ctp[r] = bb;
  const float* arow_es = edge_s + (size_t)(base + n) * 16;
#pragma unroll
  for (int k0 = 0; k0 < 16; k0 += 4)
    ctp = wmma_ab(arow_es, k0, Wtp, 16, k0, lane, ctp);

  // en = [ns[src] | ns[dst]] @ Wen + ben   (K = 32)
  int se = src[base + n], de = dst[base + n];
  v8f cen;
  bb = ben[n];
#pragma unroll
  for (int r = 0; r < 8; ++r) cen[r] = bb;
  const float* a0 = ns + (size_t)se * 16;
  const float* a1 = ns + (size_t)de * 16;
#pragma unroll
  for (int k0 = 0; k0 < 16; k0 += 4)
    cen = wmma_ab(a0, k0, Wen, 16, k0, lane, cen);
#pragma unroll
  for (int k0 = 0; k0 < 16; k0 += 4)
    cen = wmma_ab(a1, k0, Wen, 16, k0 + 16, lane, cen);

  // tm = en * tp ; gate MLP
  v8f tm;
#pragma unroll
  for (int r = 0; r < 8; ++r) tm[r] = cen[r] * ctp[r];
  c_to_lds(sm, tm, lane);
  const float* arow_l = sm + n * 18;

  v8f c1;
  bb = bg1[n];
#pragma unroll
  for (int r = 0; r < 8; ++r) c1[r] = bb;
#pragma unroll
  for (int k0 = 0; k0 < 16; k0 += 4)
    c1 = wmma_ab(arow_l, k0, Wg1, 16, k0, lane, c1);
#pragma unroll
  for (int r = 0; r < 8; ++r) c1[r] = silu1(c1[r]);
  c_to_lds(sm, c1, lane);

  v8f c2;
  bb = bg2[n];
#pragma unroll
  for (int r = 0; r < 8; ++r) c2[r] = bb;
#pragma unroll
  for (int k0 = 0; k0 < 16; k0 += 4)
    c2 = wmma_ab(arow_l, k0, Wg2, 16, k0, lane, c2);

  // cosine cutoff + per-row src/dst
  float co8[8];
  int s8[8], d8[8];
#pragma unroll
  for (int r = 0; r < 8; ++r) {
    int e = base + r + 8 * hi;
    float dd = dist[e];
    co8[r] = (dd < 10.f) ? 0.5f * (__cosf(dd * kPiOverCut) + 1.f) : 0.f;
    s8[r] = src[e];
    d8[r] = dst[e];
  }
  v8f esu;
#pragma unroll
  for (int r = 0; r < 8; ++r) esu[r] = co8[r] * sigm1(c2[r]);

  // edge_s residual out + scatter-add n_es
#pragma unroll
  for (int r = 0; r < 8; ++r) {
    int e = base + r + 8 * hi;
    es_out[(size_t)e * 16 + n] = edge_s[(size_t)e * 16 + n] + esu[r];
    unsafeAtomicAdd(&n_es[(size_t)d8[r] * 16 + n], esu[r]);
  }

  // vc = es_upd @ Wtv + btv  (three 16x16 column tiles: tc, ec, rc)
  c_to_lds(sm, esu, lane);
  v8f ct, ce, cr;
  bb = btv[n];
#pragma unroll
  for (int r = 0; r < 8; ++r) ct[r] = bb;
  bb = btv[16 + n];
#pragma unroll
  for (int r = 0; r < 8; ++r) ce[r] = bb;
  bb = btv[32 + n];
#pragma unroll
  for (int r = 0; r < 8; ++r) cr[r] = bb;
#pragma unroll
  for (int k0 = 0; k0 < 16; k0 += 4)
    ct = wmma_ab(arow_l, k0, Wtv, 48, k0, lane, ct);
#pragma unroll
  for (int k0 = 0; k0 < 16; k0 += 4)
    ce = wmma_ab(arow_l, k0, Wtv + 16, 48, k0, lane, ce);
#pragma unroll
  for (int k0 = 0; k0 < 16; k0 += 4)
    cr = wmma_ab(arow_l, k0, Wtv + 32, 48, k0, lane, cr);

  // ev_upd, edge_v residual out, scatter-add n_ev
#pragma unroll
  for (int d = 0; d < 3; ++d) {
#pragma unroll
    for (int r = 0; r < 8; ++r) {
      int e = base + r + 8 * hi;
      size_t ofs = ((size_t)e * 3 + d) * 16 + n;
      float ev = edge_v[ofs];
      float nvs = nv[((size_t)s8[r] * 3 + d) * 16 + n];
      float vn = vctr[(size_t)e * 3 + d];
      float upd = (ev * ct[r] + nvs * ce[r] + vn * cr[r]) * co8[r];
      ev_out[ofs] = ev + upd;
      unsafeAtomicAdd(&n_ev[((size_t)d8[r] * 3 + d) * 16 + n], upd);
    }
  }
}

// ---------------------------------------------------------------------------
// Kernel 3: node output projections + residuals + LayerNorm + CoorsNorm
// ---------------------------------------------------------------------------
__global__ void node_out_kernel(
    const float* __restrict__ node_s, const float* __restrict__ node_v,
    const float* __restrict__ n_es, const float* __restrict__ n_ev,
    const float* __restrict__ nvn, const float* __restrict__ Wonv,
    const float* __restrict__ Wo1, const float* __restrict__ bo1,
    const float* __restrict__ Wo2, const float* __restrict__ bo2,
    const float* __restrict__ ln_g, const float* __restrict__ ln_b,
    const float* __restrict__ cn_scale, float* __restrict__ o_ns,
    float* __restrict__ o_nv) {
  __shared__ float smem[8][16 * 18];
  int wave = threadIdx.x >> 5;
  int lane = threadIdx.x & 31;
  int tile = blockIdx.x * (blockDim.x >> 5) + wave;
  if (tile >= kNodeTiles) return;
  int base = tile * 16;
  int n = lane & 15, hi = lane >> 4;
  float* sm = smem[wave];

  // h = silu([n_es | nv_norm] @ Wo1 + bo1)   (K = 32)
  v8f ch;
  float bb = bo1[n];
#pragma unroll
  for (int r = 0; r < 8; ++r) ch[r] = bb;
  const float* aes = n_es + (size_t)(base + n) * 16;
  const float* anv = nvn + (size_t)(base + n) * 16;
#pragma unroll
  for (int k0 = 0; k0 < 16; k0 += 4)
    ch = wmma_ab(aes, k0, Wo1, 16, k0, lane, ch);
#pragma unroll
  for (int k0 = 0; k0 < 16; k0 += 4)
    ch = wmma_ab(anv, k0, Wo1, 16, k0 + 16, lane, ch);
#pragma unroll
  for (int r = 0; r < 8; ++r) ch[r] = silu1(ch[r]);
  c_to_lds(sm, ch, lane);
  const float* arow_l = sm + n * 18;

  // node_s_out = node_s + h @ Wo2 + bo2, then LayerNorm over 128 features
  float sv[8][8];
  float sum[8], sq[8];
#pragma unroll
  for (int r = 0; r < 8; ++r) { sum[r] = 0.f; sq[r] = 0.f; }
#pragma unroll
  for (int t = 0; t < 8; ++t) {
    v8f c;
    bb = bo2[t * 16 + n];
#pragma unroll
    for (int r = 0; r < 8; ++r) c[r] = bb;
#pragma unroll
    for (int k0 = 0; k0 < 16; k0 += 4)
      c = wmma_ab(arow_l, k0, Wo2 + t * 16, 128, k0, lane, c);
#pragma unroll
    for (int r = 0; r < 8; ++r) {
      float x = node_s[(size_t)(base + r + 8 * hi) * 128 + t * 16 + n] + c[r];
      sv[t][r] = x;
      sum[r] += x;
      sq[r] += x * x;
    }
  }
  // reduce across the 16 lanes of each half (xor masks < 16 stay in-half)
#pragma unroll
  for (int m = 1; m < 16; m <<= 1) {
#pragma unroll
    for (int r = 0; r < 8; ++r) {
      sum[r] += __shfl_xor(sum[r], m, 32);
      sq[r] += __shfl_xor(sq[r], m, 32);
    }
  }
  float mu[8], inv[8];
#pragma unroll
  for (int r = 0; r < 8; ++r) {
    mu[r] = sum[r] * (1.f / 128.f);
    float var = sq[r] * (1.f / 128.f) - mu[r] * mu[r];
    inv[r] = rsqrtf(var + 1e-5f);
  }
#pragma unroll
  for (int t = 0; t < 8; ++t) {
    float g = ln_g[t * 16 + n], b2 = ln_b[t * 16 + n];
#pragma unroll
    for (int r = 0; r < 8; ++r)
      o_ns[(size_t)(base + r + 8 * hi) * 128 + t * 16 + n] =
          (sv[t][r] - mu[r]) * inv[r] * g + b2;
  }

  // node_v_out = CoorsNorm(node_v + n_ev @ Wonv) * cn_scale
#pragma unroll
  for (int t = 0; t < 8; ++t) {
    float vt[3][8];
#pragma unroll
    for (int d = 0; d < 3; ++d) {
      v8f c;
#pragma unroll
      for (int r = 0; r < 8; ++r) c[r] = 0.f;
      const float* aev = n_ev + ((size_t)(base + n) * 3 + d) * 16;
#pragma unroll
      for (int k0 = 0; k0 < 16; k0 += 4)
        c = wmma_ab(aev, k0, Wonv + t * 16, 128, k0, lane, c);
#pragma unroll
      for (int r = 0; r < 8; ++r)
        vt[d][r] =
            node_v[((size_t)(base + r + 8 * hi) * 3 + d) * 128 + t * 16 + n] +
            c[r];
    }
    float cs = cn_scale[t * 16 + n];
#pragma unroll
    for (int r = 0; r < 8; ++r) {
      float nrm = sqrtf(vt[0][r] * vt[0][r] + vt[1][r] * vt[1][r] +
                        vt[2][r] * vt[2][r]);
      float scl = cs * frcp(fmaxf(nrm, 1e-8f));
#pragma unroll
      for (int d = 0; d < 3; ++d)
        o_nv[((size_t)(base + r + 8 * hi) * 3 + d) * 128 + t * 16 + n] =
            vt[d][r] * scl;
    }
  }
}

// ---------------------------------------------------------------------------
extern "C" void kernel_launch(void* const* d_in, const int* in_sizes, int n_in,
                              void* d_out, int out_size, void* d_ws,
                              size_t ws_size, hipStream_t stream) {
  const float* node_s = (const float*)d_in[0];
  const float* node_v = (const float*)d_in[1];
  const float* edge_s = (const float*)d_in[2];
  const float* edge_v = (const float*)d_in[3];
  const float* dist = (const float*)d_in[4];
  const float* vctr = (const float*)d_in[5];
  const int* src = (const int*)d_in[6];
  const int* dst = (const int*)d_in[7];
  const float* Wns = (const float*)d_in[8];
  const float* bns = (const float*)d_in[9];
  const float* Wnv = (const float*)d_in[10];
  const float* Wen = (const float*)d_in[11];
  const float* ben = (const float*)d_in[12];
  const float* Wtp = (const float*)d_in[13];
  const float* btp = (const float*)d_in[14];
  const float* Wg1 = (const float*)d_in[15];
  const float* bg1 = (const float*)d_in[16];
  const float* Wg2 = (const float*)d_in[17];
  const float* bg2 = (const float*)d_in[18];
  const float* Wtv = (const float*)d_in[19];
  const float* btv = (const float*)d_in[20];
  const float* Wonv = (const float*)d_in[21];
  const float* Wo1 = (const float*)d_in[22];
  const float* bo1 = (const float*)d_in[23];
  const float* Wo2 = (const float*)d_in[24];
  const float* bo2 = (const float*)d_in[25];
  const float* ln_g = (const float*)d_in[26];
  const float* ln_b = (const float*)d_in[27];
  const float* cn_scale = (const float*)d_in[28];

  // workspace layout (floats)
  float* ws = (float*)d_ws;
  float* ns = ws;                     // N*16
  float* nv = ns + (size_t)kN * 16;   // N*48
  float* nvn = nv + (size_t)kN * 48;  // N*16
  float* n_es = nvn + (size_t)kN * 16;  // N*16  (contiguous with n_ev)
  float* n_ev = n_es + (size_t)kN * 16; // N*48

  // output layout (tuple order, flat)
  float* out = (float*)d_out;
  float* o_ns = out;                          // N*128
  float* o_nv = o_ns + (size_t)kN * 128;      // N*384
  float* o_es = o_nv + (size_t)kN * 384;      // E*16
  float* o_ev = o_es + (size_t)kE * 16;       // E*48

  // zero scatter accumulators (n_es + n_ev are contiguous: N*64 floats)
  hipMemsetAsync(n_es, 0, (size_t)kN * 64 * sizeof(float), stream);

  dim3 blk(256);
  int node_blocks = (kNodeTiles + 7) / 8;  // 8 waves per block, 16 rows/wave
  int edge_blocks = (kEdgeTiles + 7) / 8;

  node_in_kernel<<<node_blocks, blk, 0, stream>>>(node_s, node_v, Wns, bns,
                                                  Wnv, ns, nv, nvn);
  edge_kernel<<<edge_blocks, blk, 0, stream>>>(
      edge_s, edge_v, dist, vctr, src, dst, Wen, ben, Wtp, btp, Wg1, bg1, Wg2,
      bg2, Wtv, btv, ns, nv, n_es, n_ev, o_es, o_ev);
  node_out_kernel<<<node_blocks, blk, 0, stream>>>(
      node_s, node_v, n_es, n_ev, nvn, Wonv, Wo1, bo1, Wo2, bo2, ln_g, ln_b,
      cn_scale, o_ns, o_nv);
}